// GraphSAGE_24541443129509
// MI455X (gfx1250) — compile-verified
//
#include <hip/hip_runtime.h>
#include <math.h>

#define NNODES 50000
#define NEDGES 800000
#define D 128
#define DOUT 32

typedef __attribute__((ext_vector_type(2))) float v2f;
typedef __attribute__((ext_vector_type(8))) float v8f;

__device__ __forceinline__ float elu1(float v) { return v > 0.0f ? v : expm1f(v); }

// ---- degree / inverse-degree -------------------------------------------------
__global__ void k_degree(const long long* __restrict__ dst, float* __restrict__ deg, int ne) {
    int e = blockIdx.x * blockDim.x + threadIdx.x;
    if (e < ne) unsafeAtomicAdd(&deg[(int)dst[e]], 1.0f);
}

__global__ void k_invdeg(float* __restrict__ deg, int n) {
    int i = blockIdx.x * blockDim.x + threadIdx.x;
    if (i < n) deg[i] = 1.0f / fmaxf(deg[i], 1.0f);
}

// ---- scatter-add: one wave per edge, float4 per lane (512B coalesced row) ----
__global__ void k_scatter(const float* __restrict__ x, const long long* __restrict__ src,
                          const long long* __restrict__ dst, float* __restrict__ agg, int ne) {
    int wid = (blockIdx.x * blockDim.x + threadIdx.x) >> 5;
    if (wid >= ne) return;
    int lane = threadIdx.x & 31;
    int s = (int)src[wid];
    int d = (int)dst[wid];
    float4 v = ((const float4*)(x + (size_t)s * D))[lane];
    float* ap = agg + (size_t)d * D + lane * 4;
    unsafeAtomicAdd(ap + 0, v.x);
    unsafeAtomicAdd(ap + 1, v.y);
    unsafeAtomicAdd(ap + 2, v.z);
    unsafeAtomicAdd(ap + 3, v.w);
}

// ---- fused GEMM:  out = act( scale(A) @ W1 [+ X2 @ W2] + bias ) --------------
// One wave per 16x16 output tile; block covers 16 rows x NCOLS cols.
// A staged in LDS with stride 132 (bank-conflict-free for 64x4B banks).
template <int NCOLS, bool FUSE, bool ACT>
__global__ __launch_bounds__(NCOLS * 2)
void k_gemm(const float* __restrict__ A,      // [N,128] (agg if FUSE, else features)
            const float* __restrict__ X2,     // [N,128] root features (FUSE only)
            const float* __restrict__ invdeg, // [N]      (FUSE only)
            const float* __restrict__ W1,     // [128,NCOLS]
            const float* __restrict__ W2,     // [128,NCOLS] (FUSE only)
            const float* __restrict__ bias,   // [NCOLS]
            float* __restrict__ out)          // [N,NCOLS]
{
    constexpr int NW = NCOLS / 16;            // waves per block
    __shared__ float As[16 * 132];
    __shared__ float Xs[FUSE ? 16 * 132 : 1];

    const int row0 = blockIdx.x * 16;
    const int tid  = threadIdx.x;

    // cooperative, coalesced stage of the A (and X2) row tile
    for (int idx = tid; idx < 16 * 128; idx += NW * 32) {
        int r = idx >> 7, c = idx & 127;
        float sc = FUSE ? invdeg[row0 + r] : 1.0f;
        As[r * 132 + c] = A[(size_t)(row0 + r) * D + c] * sc;
        if (FUSE) Xs[r * 132 + c] = X2[(size_t)(row0 + r) * D + c];
    }
    __syncthreads();

    const int lane = tid & 31;
    const int wave = tid >> 5;
    const int n    = wave * 16 + (lane & 15);   // output column (global within NCOLS)
    const int mrow = lane & 15;                 // A row within tile
    const int koff = (lane >> 4) * 2;           // K sub-offset per 16x16x4 layout

    v8f acc = {};
#pragma unroll
    for (int k = 0; k < D; k += 4) {
        v2f a, b;
        a.x = As[mrow * 132 + k + koff];
        a.y = As[mrow * 132 + k + koff + 1];
        b.x = W1[(size_t)(k + koff) * NCOLS + n];
        b.y = W1[(size_t)(k + koff + 1) * NCOLS + n];
        acc = __builtin_amdgcn_wmma_f32_16x16x4_f32(false, a, false, b, (short)0, acc, false, false);
        if (FUSE) {
            v2f a2, b2;
            a2.x = Xs[mrow * 132 + k + koff];
            a2.y = Xs[mrow * 132 + k + koff + 1];
            b2.x = W2[(size_t)(k + koff) * NCOLS + n];
            b2.y = W2[(size_t)(k + koff + 1) * NCOLS + n];
            acc = __builtin_amdgcn_wmma_f32_16x16x4_f32(false, a2, false, b2, (short)0, acc, false, false);
        }
    }

    const float bv = bias[n];
    const int rbase = row0 + ((lane >> 4) * 8);   // C/D layout: vgpr r -> row r (+8 for hi lanes)
#pragma unroll
    for (int r = 0; r < 8; ++r) {
        float v = acc[r] + bv;
        if (ACT) v = elu1(v);
        out[(size_t)(rbase + r) * NCOLS + n] = v;
    }
}

// -----------------------------------------------------------------------------
extern "C" void kernel_launch(void* const* d_in, const int* in_sizes, int n_in,
                              void* d_out, int out_size, void* d_ws, size_t ws_size,
                              hipStream_t stream) {
    const float*     x         = (const float*)d_in[0];
    const long long* ei        = (const long long*)d_in[1];
    const float*     conv_Wl   = (const float*)d_in[2];
    const float*     conv_bl   = (const float*)d_in[3];
    const float*     conv_Wr   = (const float*)d_in[4];
    const float*     lin_W     = (const float*)d_in[5];
    const float*     lin_b     = (const float*)d_in[6];
    const float*     lin_W_out = (const float*)d_in[7];
    const float*     lin_b_out = (const float*)d_in[8];

    const long long* src = ei;
    const long long* dst = ei + NEDGES;

    // workspace layout
    char*  ws   = (char*)d_ws;
    float* deg  = (float*)ws;                                          // 50000 f32 (pad to 204800 B)
    float* agg  = (float*)(ws + 204800);                               // 25.6 MB
    float* buf0 = (float*)(ws + 204800 + (size_t)NNODES * D * 4);      // 25.6 MB
    float* buf1 = (float*)(ws + 204800 + 2 * (size_t)NNODES * D * 4);  // 25.6 MB
    float* bufs[2] = {buf0, buf1};

    // degrees
    hipMemsetAsync(deg, 0, NNODES * sizeof(float), stream);
    k_degree<<<(NEDGES + 255) / 256, 256, 0, stream>>>(dst, deg, NEDGES);
    k_invdeg<<<(NNODES + 255) / 256, 256, 0, stream>>>(deg, NNODES);

    // 3 SAGEConv layers
    const float* cur = x;
    for (int i = 0; i < 3; ++i) {
        hipMemsetAsync(agg, 0, (size_t)NNODES * D * sizeof(float), stream);
        k_scatter<<<((size_t)NEDGES * 32 + 255) / 256, 256, 0, stream>>>(cur, src, dst, agg, NEDGES);
        float* o = bufs[i & 1];
        k_gemm<128, true, true><<<NNODES / 16, 256, 0, stream>>>(
            agg, cur, deg,
            conv_Wl + (size_t)i * D * D, conv_Wr + (size_t)i * D * D,
            conv_bl + (size_t)i * D, o);
        cur = o;
    }
    // cur == buf0 here

    // 2 hidden linears
    for (int i = 0; i < 2; ++i) {
        float* o = bufs[(i + 1) & 1];
        k_gemm<128, false, true><<<NNODES / 16, 256, 0, stream>>>(
            cur, nullptr, nullptr,
            lin_W + (size_t)i * D * D, nullptr, lin_b + (size_t)i * D, o);
        cur = o;
    }
    // cur == buf0

    // output head (no activation)
    k_gemm<32, false, false><<<NNODES / 16, 64, 0, stream>>>(
        cur, nullptr, nullptr, lin_W_out, nullptr, lin_b_out, (float*)d_out);
}